// GCN_3layer_plain_67972152427190
// MI455X (gfx1250) — compile-verified
//
#include <hip/hip_runtime.h>

// GCN 3-layer, MI455X (gfx1250, wave32).
// GEMMs: V_WMMA_F32_16X16X4_F32, A-tile + transposed-W staged in LDS with
// coalesced b128 global loads; fragments fed by bank-conflict-free ds_load_b64.
// Edge scatter is the roofline term (~2.5 GB of L2-resident gather+atomic traffic).

typedef __attribute__((ext_vector_type(2))) float v2f;
typedef __attribute__((ext_vector_type(8))) float v8f;

#define NODE_DIM 64
#define NUM_GRAPHS 256

__global__ void zero_f32(float* __restrict__ p, size_t n) {
  size_t i = (size_t)blockIdx.x * blockDim.x + threadIdx.x;
  if (i < n) p[i] = 0.0f;
}

// Block = 128 threads = 4 waves; block computes a 64-row x 64-col tile of M = H @ W.
// Wave w owns rows [w*16, w*16+16). LDS row stride K+2 avoids bank conflicts
// on the strided fragment reads while keeping 8B alignment for b64 LDS loads.
template <int K, bool RELU>
__global__ __launch_bounds__(128)
void gemm_wmma_f32(const float* __restrict__ H, const float* __restrict__ W,
                   float* __restrict__ M, int n_rows) {
  constexpr int Kp = K + 2;
  extern __shared__ float smem[];
  float* sA  = smem;             // [64 rows][Kp]
  float* sWt = smem + 64 * Kp;   // [64 cols][Kp]  (W transposed)

  const int tid  = threadIdx.x;
  const int row0 = blockIdx.x * 64;

  // Stage W transposed: sWt[c][k] = W[k][c] (coalesced global read).
  for (int i = tid; i < K * 64; i += 128) {
    int k = i >> 6;
    int c = i & 63;
    sWt[c * Kp + k] = W[i];
  }

  // Stage A tile (64 rows x K): coalesced float4 reads, zero-fill OOB rows,
  // ReLU fused here so the WMMA loop is divergence-free.
  for (int i = tid * 4; i < 64 * K; i += 128 * 4) {
    int r = i / K;                 // K is 64/128 -> shift
    int k = i & (K - 1);
    float4 v = {0.0f, 0.0f, 0.0f, 0.0f};
    int gr = row0 + r;
    if (gr < n_rows) v = *(const float4*)(H + (size_t)gr * K + k);
    if (RELU) {
      v.x = fmaxf(v.x, 0.0f); v.y = fmaxf(v.y, 0.0f);
      v.z = fmaxf(v.z, 0.0f); v.w = fmaxf(v.w, 0.0f);
    }
    float* d = &sA[r * Kp + k];    // (r*Kp+k) even -> 8B aligned
    d[0] = v.x; d[1] = v.y; d[2] = v.z; d[3] = v.w;
  }
  __syncthreads();

  const int wave  = tid >> 5;
  const int lane  = tid & 31;
  const int n     = lane & 15;   // col within 16-block / row within A tile
  const int khalf = lane >> 4;   // 0: K{0,1}, 1: K{2,3} of each K=4 step
  const float* arow = &sA[(wave * 16 + n) * Kp];

  v8f acc0 = {}; v8f acc1 = {}; v8f acc2 = {}; v8f acc3 = {};

  for (int k0 = 0; k0 < K; k0 += 4) {
    const int ka = k0 + 2 * khalf;
    // A fragment: lane holds (row, K=ka),(row, K=ka+1)  -> one ds_load_b64
    v2f a  = *(const v2f*)(arow + ka);
    // B fragments: lane holds (K=ka, col),(K=ka+1, col) -> b64 from sWt
    v2f b0 = *(const v2f*)&sWt[(n +  0) * Kp + ka];
    v2f b1 = *(const v2f*)&sWt[(n + 16) * Kp + ka];
    v2f b2 = *(const v2f*)&sWt[(n + 32) * Kp + ka];
    v2f b3 = *(const v2f*)&sWt[(n + 48) * Kp + ka];

    acc0 = __builtin_amdgcn_wmma_f32_16x16x4_f32(false, a, false, b0, (short)0, acc0, false, false);
    acc1 = __builtin_amdgcn_wmma_f32_16x16x4_f32(false, a, false, b1, (short)0, acc1, false, false);
    acc2 = __builtin_amdgcn_wmma_f32_16x16x4_f32(false, a, false, b2, (short)0, acc2, false, false);
    acc3 = __builtin_amdgcn_wmma_f32_16x16x4_f32(false, a, false, b3, (short)0, acc3, false, false);
  }

  // C/D layout: VGPR v -> M = v + 8*(lane/16), N = lane%16
  const int row_base = row0 + wave * 16;
  for (int v = 0; v < 8; ++v) {
    int r = row_base + v + 8 * khalf;
    if (r < n_rows) {
      float* op = M + (size_t)r * NODE_DIM + n;
      op[0]  = acc0[v];
      op[16] = acc1[v];
      op[32] = acc2[v];
      op[48] = acc3[v];
    }
  }
}

// 8 threads per edge, 8 channels each: coalesced 256B gather + f32 atomics.
__global__ __launch_bounds__(256)
void scatter_add_edges(const float* __restrict__ Msg,
                       const long long* __restrict__ src,
                       const long long* __restrict__ dst,
                       float* __restrict__ Hout, int n_edges) {
  int t = blockIdx.x * blockDim.x + threadIdx.x;
  int edge = t >> 3;
  int part = t & 7;
  if (edge >= n_edges) return;
  int s = (int)src[edge];
  int d = (int)dst[edge];
  const float4* mp = (const float4*)(Msg + (size_t)s * NODE_DIM + part * 8);
  float4 v0 = mp[0];
  float4 v1 = mp[1];
  float* hp = Hout + (size_t)d * NODE_DIM + part * 8;
  atomicAdd(hp + 0, v0.x);
  atomicAdd(hp + 1, v0.y);
  atomicAdd(hp + 2, v0.z);
  atomicAdd(hp + 3, v0.w);
  atomicAdd(hp + 4, v1.x);
  atomicAdd(hp + 5, v1.y);
  atomicAdd(hp + 6, v1.z);
  atomicAdd(hp + 7, v1.w);
}

// Per-graph sums and counts via atomics (batch is sorted -> atomics mostly local).
__global__ __launch_bounds__(256)
void pool_accum(const float* __restrict__ H, const long long* __restrict__ batch,
                float* __restrict__ sums, float* __restrict__ counts, int n_nodes) {
  int t = blockIdx.x * blockDim.x + threadIdx.x;
  int node = t >> 3;
  int part = t & 7;
  if (node >= n_nodes) return;
  int g = (int)batch[node];
  const float* hp = H + (size_t)node * NODE_DIM + part * 8;
  float* sp = sums + (size_t)g * NODE_DIM + part * 8;
  for (int i = 0; i < 8; ++i) atomicAdd(sp + i, hp[i]);
  if (part == 0) atomicAdd(counts + g, 1.0f);
}

// out[g][j] = (sums[g]/max(count,1)) . Wlin[:,j]   (256x16, trivial)
__global__ __launch_bounds__(256)
void final_project(const float* __restrict__ sums, const float* __restrict__ counts,
                   const float* __restrict__ Wlin, float* __restrict__ out) {
  int t = blockIdx.x * blockDim.x + threadIdx.x;
  int g = t >> 4;
  int j = t & 15;
  if (g >= NUM_GRAPHS) return;
  float inv = 1.0f / fmaxf(counts[g], 1.0f);
  float acc = 0.0f;
  for (int c = 0; c < NODE_DIM; ++c)
    acc += sums[g * NODE_DIM + c] * Wlin[c * 16 + j];
  out[g * 16 + j] = acc * inv;
}

extern "C" void kernel_launch(void* const* d_in, const int* in_sizes, int n_in,
                              void* d_out, int out_size, void* d_ws, size_t ws_size,
                              hipStream_t stream) {
  const float* x        = (const float*)d_in[0];
  const float* W1       = (const float*)d_in[1];
  const float* W2       = (const float*)d_in[2];
  const float* W3       = (const float*)d_in[3];
  const float* Wlin     = (const float*)d_in[4];
  const long long* ei   = (const long long*)d_in[5];
  const long long* batc = (const long long*)d_in[6];
  float* out = (float*)d_out;

  const int N  = in_sizes[6];                 // 100000 nodes
  const int E  = in_sizes[5] / 2;             // 1.6M edges
  const int K1 = in_sizes[1] / NODE_DIM;      // 128
  const long long* srcp = ei;
  const long long* dstp = ei + E;

  const size_t nodeElems = (size_t)N * NODE_DIM;
  float* m      = (float*)d_ws;               // messages
  float* hA     = m + nodeElems;              // node buffer A
  float* hB     = hA + nodeElems;             // node buffer B
  float* sums   = hB + nodeElems;             // [256 x 64]
  float* counts = sums + (size_t)NUM_GRAPHS * NODE_DIM;  // [256]

  const dim3 blk256(256);
  const int gridZ    = (int)((nodeElems + 255) / 256);
  const int gemmGrid = (N + 63) / 64;
  const int scatGrid = (int)(((size_t)E * 8 + 255) / 256);
  const int poolGrid = (int)(((size_t)N * 8 + 255) / 256);

  const size_t shmem128 = 2u * 64u * (128 + 2) * sizeof(float);  // 66.5 KB
  const size_t shmem64  = 2u * 64u * (64 + 2)  * sizeof(float);  // 33.8 KB

  // Layer 1: h1 = scatter(x @ W1)
  hipLaunchKernelGGL(zero_f32, dim3(gridZ), blk256, 0, stream, hA, nodeElems);
  if (K1 == 128) {
    hipLaunchKernelGGL((gemm_wmma_f32<128, false>), dim3(gemmGrid), dim3(128), shmem128, stream,
                       x, W1, m, N);
  } else {
    hipLaunchKernelGGL((gemm_wmma_f32<64, false>), dim3(gemmGrid), dim3(128), shmem64, stream,
                       x, W1, m, N);
  }
  hipLaunchKernelGGL(scatter_add_edges, dim3(scatGrid), blk256, 0, stream, m, srcp, dstp, hA, E);

  // Layer 2: h2 = scatter(relu(h1) @ W2)   (relu fused into A-tile staging)
  hipLaunchKernelGGL((gemm_wmma_f32<64, true>), dim3(gemmGrid), dim3(128), shmem64, stream,
                     hA, W2, m, N);
  hipLaunchKernelGGL(zero_f32, dim3(gridZ), blk256, 0, stream, hB, nodeElems);
  hipLaunchKernelGGL(scatter_add_edges, dim3(scatGrid), blk256, 0, stream, m, srcp, dstp, hB, E);

  // Layer 3: h3 = scatter(relu(h2) @ W3)
  hipLaunchKernelGGL((gemm_wmma_f32<64, true>), dim3(gemmGrid), dim3(128), shmem64, stream,
                     hB, W3, m, N);
  hipLaunchKernelGGL(zero_f32, dim3(gridZ), blk256, 0, stream, hA, nodeElems);
  hipLaunchKernelGGL(scatter_add_edges, dim3(scatGrid), blk256, 0, stream, m, srcp, dstp, hA, E);

  // Mean pool + linear head
  const size_t poolElems = (size_t)NUM_GRAPHS * NODE_DIM + NUM_GRAPHS;  // sums + counts
  hipLaunchKernelGGL(zero_f32, dim3((int)((poolElems + 255) / 256)), blk256, 0, stream, sums, poolElems);
  hipLaunchKernelGGL(pool_accum, dim3(poolGrid), blk256, 0, stream, hA, batc, sums, counts, N);
  hipLaunchKernelGGL(final_project, dim3((NUM_GRAPHS * 16 + 255) / 256), blk256, 0, stream,
                     sums, counts, Wlin, out);
}